// Sylo_32856499814782
// MI455X (gfx1250) — compile-verified
//
#include <hip/hip_runtime.h>

// Sylo forward, MI455X (gfx1250, wave32).
// T[o,c] = L L^T is rank-4 symmetric; a 16x16 T-tile == one V_WMMA_F32_16X16X4_F32.
// out = rs[h] + cs[w] - sum_c X.*T + bias, diagonal zeroed.

#define DIM  512
#define NB   8
#define CIN  4
#define COUT 64

typedef float v2f __attribute__((ext_vector_type(2)));
typedef float v8f __attribute__((ext_vector_type(8)));

__device__ __forceinline__ float dot4(float4 a, float4 b) {
  return fmaf(a.x, b.x, fmaf(a.y, b.y, fmaf(a.z, b.z, a.w * b.w)));
}

// ---------------------------------------------------------------------------
// rs[n,o,h] = sum_{c,w} X[n,c,h,w] * T[o,c,h,w],  T = dot4(Lh, Lw)
// block = (o, 32-row strip); 8 waves x 4 rows; lane owns 2 w (float2 X loads
// -> v_pk_fma_f32); Lw LDS reads hoisted across the 4 rows.
// ---------------------------------------------------------------------------
__global__ __launch_bounds__(256) void sylo_rowsums(const float* __restrict__ X,
                                                    const float* __restrict__ L,
                                                    float* __restrict__ rs) {
  __shared__ float4 Ls4[CIN * DIM];                // L[o] : 32 KB
  const int o = blockIdx.x, strip = blockIdx.y;
  const int tid = threadIdx.x;
  const float4* Lg = (const float4*)(L + (size_t)o * CIN * DIM * 4);
  for (int i = tid; i < CIN * DIM; i += 256) Ls4[i] = Lg[i];
  __syncthreads();

  const int wave = tid >> 5, lane = tid & 31;
  const int hbase = strip * 32 + wave * 4;

  float2 racc2[4][NB];                             // [row][n], {even w, odd w}
#pragma unroll
  for (int rr = 0; rr < 4; ++rr)
#pragma unroll
    for (int n = 0; n < NB; ++n) racc2[rr][n] = make_float2(0.f, 0.f);

  for (int wc = 0; wc < 8; ++wc) {
    const int w2 = wc * 64 + lane * 2;
#pragma unroll
    for (int c = 0; c < CIN; ++c) {
      const float4 lw0 = Ls4[c * DIM + w2 + 0];    // reused across 4 rows
      const float4 lw1 = Ls4[c * DIM + w2 + 1];
#pragma unroll
      for (int rr = 0; rr < 4; ++rr) {
        const int h = hbase + rr;
        const float4 lh = Ls4[c * DIM + h];        // broadcast read
        const float t0 = dot4(lh, lw0);
        const float t1 = dot4(lh, lw1);
#pragma unroll
        for (int n = 0; n < NB; ++n) {
          const float2 x = *(const float2*)(X + (((size_t)n * CIN + c) * DIM + h) * DIM + w2);
          racc2[rr][n].x = fmaf(x.x, t0, racc2[rr][n].x);
          racc2[rr][n].y = fmaf(x.y, t1, racc2[rr][n].y);
        }
      }
    }
  }
#pragma unroll
  for (int rr = 0; rr < 4; ++rr)
#pragma unroll
    for (int n = 0; n < NB; ++n) {
      float v = racc2[rr][n].x + racc2[rr][n].y;
#pragma unroll
      for (int off = 16; off > 0; off >>= 1) v += __shfl_xor(v, off, 32);
      if (lane == 0) rs[(n * COUT + o) * DIM + hbase + rr] = v;
    }
}

// ---------------------------------------------------------------------------
// cs[n,o,w] = sum_{c,h} X[n,c,h,w] * T[o,c,h,w]
// block = (o, 128-col quarter); lane owns 4 w (float4 X loads); waves split h;
// deterministic LDS partial reduction (no atomics).
// ---------------------------------------------------------------------------
__global__ __launch_bounds__(256) void sylo_colsums(const float* __restrict__ X,
                                                    const float* __restrict__ L,
                                                    float* __restrict__ cs) {
  __shared__ float4 Ls4[CIN * DIM];                // 32 KB
  __shared__ float csl[8][NB][128];                // 32 KB wave partials
  const int o = blockIdx.x, wq = blockIdx.y;
  const int tid = threadIdx.x;
  const float4* Lg = (const float4*)(L + (size_t)o * CIN * DIM * 4);
  for (int i = tid; i < CIN * DIM; i += 256) Ls4[i] = Lg[i];
  __syncthreads();

  const int wave = tid >> 5, lane = tid & 31;
  const int wb = wq * 128 + lane * 4;
  const int hb = wave * 64;
  float cacc[NB][4];
#pragma unroll
  for (int n = 0; n < NB; ++n)
#pragma unroll
    for (int j = 0; j < 4; ++j) cacc[n][j] = 0.f;

#pragma unroll
  for (int c = 0; c < CIN; ++c) {
    const float4 lw0 = Ls4[c * DIM + wb + 0];
    const float4 lw1 = Ls4[c * DIM + wb + 1];
    const float4 lw2 = Ls4[c * DIM + wb + 2];
    const float4 lw3 = Ls4[c * DIM + wb + 3];
    for (int hh = 0; hh < 64; ++hh) {
      const int h = hb + hh;
      const float4 lhv = Ls4[c * DIM + h];
      const float t0 = dot4(lhv, lw0);
      const float t1 = dot4(lhv, lw1);
      const float t2 = dot4(lhv, lw2);
      const float t3 = dot4(lhv, lw3);
#pragma unroll
      for (int n = 0; n < NB; ++n) {
        const float4 x = *(const float4*)(X + (((size_t)n * CIN + c) * DIM + h) * DIM + wb);
        cacc[n][0] = fmaf(x.x, t0, cacc[n][0]);
        cacc[n][1] = fmaf(x.y, t1, cacc[n][1]);
        cacc[n][2] = fmaf(x.z, t2, cacc[n][2]);
        cacc[n][3] = fmaf(x.w, t3, cacc[n][3]);
      }
    }
  }
#pragma unroll
  for (int n = 0; n < NB; ++n) {
    float4 pv = {cacc[n][0], cacc[n][1], cacc[n][2], cacc[n][3]};
    *(float4*)&csl[wave][n][lane * 4] = pv;
  }
  __syncthreads();
  for (int i = tid; i < NB * 128; i += 256) {
    const int n = i >> 7, w = i & 127;
    float s = 0.f;
#pragma unroll
    for (int wv = 0; wv < 8; ++wv) s += csl[wv][n][w];
    cs[(n * COUT + o) * DIM + wq * 128 + w] = s;
  }
}

// ---------------------------------------------------------------------------
// Main output kernel: block = one 16x16 (h,w) tile; 8 waves x 8 o's each.
// Per (o,c): one V_WMMA_F32_16X16X4_F32 produces the T tile (K = RANK = 4),
// reused across all 8 batches. X staged in LDS as [c][n][w][h pad 20]; the
// compiler hoists the o-invariant tile values into VGPRs (>256 via
// S_SET_VGPR_MSB) and packs the accumulation into v_pk_fma_f32.
// ---------------------------------------------------------------------------
__global__ __launch_bounds__(256) void sylo_main(const float* __restrict__ X,
                                                 const float* __restrict__ L,
                                                 const float* __restrict__ bias,
                                                 const float* __restrict__ rs,
                                                 const float* __restrict__ cs,
                                                 float* __restrict__ out) {
  __shared__ __align__(16) float Xs[CIN * NB * 16 * 20];   // 40 KB
  const int h0 = blockIdx.x * 16, w0 = blockIdx.y * 16;
  const int tid = threadIdx.x;

  // Stage X tile: element (h,w) -> Xs[((c*8+n)*16 + w)*20 + h]
  for (int i = tid; i < 2048; i += 256) {
    const int q = i & 3, r = (i >> 2) & 15, n = (i >> 6) & 7, c = i >> 9;
    const float4 v = *(const float4*)(X + (((size_t)n * CIN + c) * DIM + (h0 + r)) * DIM + w0 + q * 4);
    float* dst = Xs + (c * NB + n) * 16 * 20;
    dst[(q * 4 + 0) * 20 + r] = v.x;
    dst[(q * 4 + 1) * 20 + r] = v.y;
    dst[(q * 4 + 2) * 20 + r] = v.z;
    dst[(q * 4 + 3) * 20 + r] = v.w;
  }
  __syncthreads();

  const int wave = tid >> 5, lane = tid & 31;
  const int m = lane & 15, khalf = lane >> 4;

#pragma unroll 1
  for (int oi = 0; oi < 8; ++oi) {
    const int o = wave * 8 + oi;
    const float bo = bias[o];

    // WMMA fragments. A (16x4 f32): lane m=l%16; V0 holds K0|K2, V1 holds K1|K3.
    // B (4x16) mirrors A; T symmetric => identical indexing off w0.
    v2f a[CIN], b[CIN];
#pragma unroll
    for (int c = 0; c < CIN; ++c) {
      const float* lp = L + ((size_t)(o * CIN + c) * DIM) * 4 + 2 * khalf;
      a[c] = *(const v2f*)(lp + (h0 + m) * 4);
      b[c] = *(const v2f*)(lp + (w0 + m) * 4);
    }

    float acc[NB][8];
#pragma unroll
    for (int n = 0; n < NB; ++n)
#pragma unroll
      for (int v = 0; v < 8; ++v) acc[n][v] = 0.f;

#pragma unroll
    for (int c = 0; c < CIN; ++c) {
      v8f zero = {};
      // T_tile[m_row, n_col] = sum_r L[h0+m_row, r] * L[w0+n_col, r]
      v8f t = __builtin_amdgcn_wmma_f32_16x16x4_f32(false, a[c], false, b[c],
                                                    (short)0, zero, false, false);
#pragma unroll
      for (int n = 0; n < NB; ++n) {
        const float* xp = &Xs[((c * NB + n) * 16 + m) * 20 + khalf * 8];
        const float4 x0 = *(const float4*)xp;
        const float4 x1 = *(const float4*)(xp + 4);
        acc[n][0] = fmaf(t[0], x0.x, acc[n][0]);
        acc[n][1] = fmaf(t[1], x0.y, acc[n][1]);
        acc[n][2] = fmaf(t[2], x0.z, acc[n][2]);
        acc[n][3] = fmaf(t[3], x0.w, acc[n][3]);
        acc[n][4] = fmaf(t[4], x1.x, acc[n][4]);
        acc[n][5] = fmaf(t[5], x1.y, acc[n][5]);
        acc[n][6] = fmaf(t[6], x1.z, acc[n][6]);
        acc[n][7] = fmaf(t[7], x1.w, acc[n][7]);
      }
    }

    // Epilogue: out = rs + cs - P + bias; zero diagonal.
#pragma unroll
    for (int n = 0; n < NB; ++n) {
      const int nb = (n * COUT + o) * DIM;
      const float csv = cs[nb + w0 + m];
#pragma unroll
      for (int v = 0; v < 8; ++v) {
        const int h = h0 + khalf * 8 + v;
        float val = rs[nb + h] + csv - acc[n][v] + bo;
        if (h == w0 + m) val = 0.f;
        out[((size_t)nb + h) * DIM + w0 + m] = val;
      }
    }
  }
}

// ---------------------------------------------------------------------------
extern "C" void kernel_launch(void* const* d_in, const int* in_sizes, int n_in,
                              void* d_out, int out_size, void* d_ws, size_t ws_size,
                              hipStream_t stream) {
  const float* X    = (const float*)d_in[0];   // (8,4,512,512)
  const float* L    = (const float*)d_in[1];   // (64,4,512,4)
  const float* bias = (const float*)d_in[2];   // (64,)
  float* out = (float*)d_out;                  // (8,64,512,512)

  float* rs = (float*)d_ws;                    // 8*64*512 floats (1 MB)
  float* cs = rs + NB * COUT * DIM;            // 8*64*512 floats (1 MB)

  sylo_rowsums<<<dim3(COUT, 16), 256, 0, stream>>>(X, L, rs);
  sylo_colsums<<<dim3(COUT, 4),  256, 0, stream>>>(X, L, cs);
  sylo_main   <<<dim3(32, 32),   256, 0, stream>>>(X, L, bias, rs, cs, out);
}